// MyCircuit_19945828122987
// MI455X (gfx1250) — compile-verified
//
#include <hip/hip_runtime.h>

// ---------------------------------------------------------------------------
// 22-qubit circuit: (H then RY) per qubit, CNOT ring, mean <Z_k>.
//  * fuse H+RY per qubit -> 2x2 gate F = RY * H
//  * fuse 4 flat-index bits per pass -> 16x16 Kronecker gate, applied to the
//    state (16 x 2^18 view) with chained V_WMMA_F32_16X16X4_F32 (K=16)
//  * CNOT ring is GF(2)-linear on index bits -> folded into the reduction via
//    compile-time parity masks; no permutation pass touches memory
//  * deterministic two-stage reduction (no float atomics)
// State (16 MB) lives in d_ws and stays resident in the 192 MB L2.
// Gate kernel is templated on the bit offset so all strides are immediates.
// ---------------------------------------------------------------------------

#define NQ 22
#define STATE_SIZE (1u << NQ)

typedef __attribute__((ext_vector_type(2))) float v2f;
typedef __attribute__((ext_vector_type(8))) float v8f;

struct MaskArr { unsigned m[NQ]; };

// Compile-time GF(2) composition of the CNOT ring (qubit k -> flat bit 21-k).
// m[k] is the input-bit parity mask: <Z_k> = sum_i psi_i^2 * (-1)^parity(i&m_k).
static constexpr MaskArr compute_masks() {
    MaskArr out{};
    unsigned rows[NQ] = {};
    for (int j = 0; j < NQ; ++j) rows[j] = 1u << j;             // qubit-bit space
    for (int k = 0; k < NQ; ++k) rows[(k + 1) % NQ] ^= rows[k]; // CNOT(k,k+1)
    for (int k = 0; k < NQ; ++k) {
        unsigned fm = 0;
        for (int j = 0; j < NQ; ++j)
            if ((rows[k] >> j) & 1u) fm |= 1u << (21 - j);      // qubit j -> bit 21-j
        out.m[k] = fm;
    }
    return out;
}

// ---------------------------------------------------------------------------
// Build the six 16x16 group gate matrices (row-major) from params.
// Group g covers flat bits boff[g]..boff[g]+3; group 5 (bits 18..21) uses
// identity factors on bits 18,19 (already handled by group 4).
// Gate on flat bit b acts on qubit q = 21-b:  F = RY * H.
// ---------------------------------------------------------------------------
__global__ __launch_bounds__(256) void build_matrices(
    const float* __restrict__ params, float* __restrict__ mats) {
    const int tid = threadIdx.x;          // 0..255 -> one matrix entry
    const int r = tid >> 4;
    const int cidx = tid & 15;
    const int boff[6] = {0, 4, 8, 12, 16, 18};
    const float RS2 = 0.70710678118654752f;
    for (int g = 0; g < 6; ++g) {
        float prod = 1.0f;
        for (int j = 0; j < 4; ++j) {
            const int rr = (r >> j) & 1;
            const int cc = (cidx >> j) & 1;
            float f;
            if (g == 5 && j < 2) {                 // identity factor
                f = (rr == cc) ? 1.0f : 0.0f;
            } else {
                const int bit = boff[g] + j;
                const int q = 21 - bit;
                const float th = params[q] * 0.5f;
                const float cs = cosf(th), sn = sinf(th);
                // F = RY*H: [[c-s, c+s],[c+s, s-c]] / sqrt(2)
                f = (rr == 0) ? ((cc == 0) ? (cs - sn) : (cs + sn))
                              : ((cc == 0) ? (cs + sn) : (sn - cs));
                f *= RS2;
            }
            prod *= f;
        }
        mats[g * 256 + tid] = prod;
    }
}

// |0...0> initial state (d_ws is poisoned, must be fully written)
__global__ __launch_bounds__(256) void init_state(float* __restrict__ state) {
    const unsigned g4 = blockIdx.x * blockDim.x + threadIdx.x; // one float4 each
    float4 z = make_float4(0.f, 0.f, 0.f, 0.f);
    if (g4 == 0) z.x = 1.0f;
    ((float4*)state)[g4] = z;
}

// ---------------------------------------------------------------------------
// Apply a 16x16 group gate to the whole state, in place.
// Compile-time bit offset B: tuple element c lives at base + c*(1<<B).
// A wave owns 16x16 tiles (16 tuples x 16 components = 256 disjoint floats);
// D = G * X via four chained f32 WMMAs (K=4 each), written back in place.
// All strides are immediates -> loads/stores use instruction offsets and
// vectorize for the contiguous passes (B=0, B=4).
// ---------------------------------------------------------------------------
template <int B>
__global__ __launch_bounds__(256) void apply_gate(
    float* __restrict__ state, const float* __restrict__ G) {
    const int lane = threadIdx.x & 31;
    const int wid = (blockIdx.x * blockDim.x + threadIdx.x) >> 5;
    const int lo = lane & 15;
    const int hi = lane >> 4;
    constexpr unsigned cs = 1u << B;
    constexpr unsigned lowmask = cs - 1u;

    // A-matrix (16x4 f32 layout): lanes 0-15 -> K={0,1}, lanes 16-31 -> K={2,3}
    const float* ga = G + lo * 16 + 2 * hi;
    const v2f a0 = *(const v2f*)(ga + 0);
    const v2f a1 = *(const v2f*)(ga + 4);
    const v2f a2 = *(const v2f*)(ga + 8);
    const v2f a3 = *(const v2f*)(ga + 12);

    // This lane's B/D column base, per tile t: r = (wid*4+t)*16 + lo
    const unsigned r0 = (unsigned)wid * 64u + (unsigned)lo;
    const unsigned o = (unsigned)(2 * hi) * cs;   // K-pair select for B rows
    const unsigned so = (unsigned)(8 * hi) * cs;  // M-half select for D rows

    for (int t = 0; t < 4; ++t) {
        const unsigned r = r0 + (unsigned)t * 16u;
        const unsigned base = ((r >> B) << (B + 4)) | (r & lowmask);
        float* p = state + base;

        if (t < 3) {  // prefetch next tile while WMMAs run
            const unsigned rn = r + 16u;
            const unsigned bn = ((rn >> B) << (B + 4)) | (rn & lowmask);
            __builtin_prefetch(state + bn + o, 0, 3);
        }

        // B-matrix (4x16): lane half selects K pair, N = lo
        v2f b0, b1, b2, b3;
        b0.x = p[o + 0 * cs];   b0.y = p[o + 1 * cs];
        b1.x = p[o + 4 * cs];   b1.y = p[o + 5 * cs];
        b2.x = p[o + 8 * cs];   b2.y = p[o + 9 * cs];
        b3.x = p[o + 12 * cs];  b3.y = p[o + 13 * cs];

        v8f c = {};
        c = __builtin_amdgcn_wmma_f32_16x16x4_f32(false, a0, false, b0, (short)0, c, false, false);
        c = __builtin_amdgcn_wmma_f32_16x16x4_f32(false, a1, false, b1, (short)0, c, false, false);
        c = __builtin_amdgcn_wmma_f32_16x16x4_f32(false, a2, false, b2, (short)0, c, false, false);
        c = __builtin_amdgcn_wmma_f32_16x16x4_f32(false, a3, false, b3, (short)0, c, false, false);

        // D: lane holds M = v + 8*hi, N = lo
        #pragma unroll
        for (int v = 0; v < 8; ++v) p[so + (unsigned)v * cs] = c[v];
    }
}

// Per-block deterministic partial sums of psi_i^2 * sum_k (-1)^parity(i&m_k)
__global__ __launch_bounds__(256) void reduce_kernel(
    const float* __restrict__ state, float* __restrict__ partials, MaskArr mk) {
    __shared__ float sm[256];
    const int tid = threadIdx.x;
    const unsigned gt = blockIdx.x * 256u + (unsigned)tid;
    float acc = 0.f;
    #pragma unroll 4
    for (int e = 0; e < 16; ++e) {
        const unsigned i = gt * 16u + (unsigned)e;
        const float s = state[i];
        const float p = s * s;
        int odd = 0;
        #pragma unroll
        for (int k = 0; k < NQ; ++k) odd += (__popc(i & mk.m[k]) & 1);
        acc += p * (float)(NQ - 2 * odd);
    }
    sm[tid] = acc;
    __syncthreads();
    for (int s2 = 128; s2 > 0; s2 >>= 1) {
        if (tid < s2) sm[tid] += sm[tid + s2];
        __syncthreads();
    }
    if (tid == 0) partials[blockIdx.x] = sm[0];
}

__global__ __launch_bounds__(256) void finalize_kernel(
    const float* __restrict__ partials, float* __restrict__ out) {
    __shared__ float sm[256];
    const int tid = threadIdx.x;
    sm[tid] = partials[tid] + partials[tid + 256] + partials[tid + 512] + partials[tid + 768];
    __syncthreads();
    for (int s = 128; s > 0; s >>= 1) {
        if (tid < s) sm[tid] += sm[tid + s];
        __syncthreads();
    }
    if (tid == 0) out[0] = sm[0] * (1.0f / (float)NQ);
}

extern "C" void kernel_launch(void* const* d_in, const int* in_sizes, int n_in,
                              void* d_out, int out_size, void* d_ws, size_t ws_size,
                              hipStream_t stream) {
    (void)in_sizes; (void)n_in; (void)out_size; (void)ws_size;
    const float* params = (const float*)d_in[0];     // 22 f32 angles
    float* state = (float*)d_ws;                     // 2^22 floats (16 MB)
    float* mats = state + STATE_SIZE;                // 6 * 256 floats
    float* partials = mats + 6 * 256;                // 1024 floats
    float* out = (float*)d_out;

    build_matrices<<<1, 256, 0, stream>>>(params, mats);
    init_state<<<STATE_SIZE / 1024, 256, 0, stream>>>(state);

    // 512 blocks * 8 waves * 4 tiles = 16384 tiles = 2^22 elements per pass
    apply_gate<0><<<512, 256, 0, stream>>>(state, mats + 0 * 256);
    apply_gate<4><<<512, 256, 0, stream>>>(state, mats + 1 * 256);
    apply_gate<8><<<512, 256, 0, stream>>>(state, mats + 2 * 256);
    apply_gate<12><<<512, 256, 0, stream>>>(state, mats + 3 * 256);
    apply_gate<16><<<512, 256, 0, stream>>>(state, mats + 4 * 256);
    apply_gate<18><<<512, 256, 0, stream>>>(state, mats + 5 * 256);

    constexpr MaskArr mk = compute_masks();
    reduce_kernel<<<1024, 256, 0, stream>>>(state, partials, mk);
    finalize_kernel<<<1, 256, 0, stream>>>(partials, out);
}